// LatentODE_67061619360412
// MI455X (gfx1250) — compile-verified
//
#include <hip/hip_runtime.h>
#include <math.h>

#define T_STEPS 4096
#define HID 128
#define INP 8
#define MID 50
#define KSUB 10

typedef __attribute__((ext_vector_type(2))) float v2f;
typedef __attribute__((ext_vector_type(8))) float v8f;

// Branchless tanh: 1 - 2/(exp(2v)+1). v_exp_f32 computes 2^x, so scale by
// 2*log2(e). Saturates correctly: v>>0 -> exp=inf -> rcp=0 -> 1;
// v<<0 -> exp=0 -> 1-2 = -1. No EXEC-mask branches on the serial path.
__device__ __forceinline__ float fast_tanh(float v) {
    float e = __builtin_amdgcn_exp2f(v * 2.8853900817779268f);
    return 1.0f - 2.0f * __builtin_amdgcn_rcpf(e + 1.0f);
}

// XOR-butterfly within the wave via a single ds_swizzle (group-of-32 mode:
// and_mask=0x1F, xor_mask=m). Replaces __shfl_xor's mbcnt/cmp/cndmask
// scaffolding with one DS op. Masks 1/2/4/8 never cross a 16-lane half.
#define SWIZ_XOR(v, m) \
    __int_as_float(__builtin_amdgcn_ds_swizzle(__float_as_int(v), (0x1F | ((m) << 10))))

// ---------------------------------------------------------------------------
// Phase 1: strictly sequential ODE scan. One workgroup, 128 threads (4 waves
// on one WGP, 1 wave/SIMD). Weight COLUMNS live in registers (loop-invariant
// across all 40960 substeps); only activations (sz, sh) go through LDS.
// Two barriers per substep is the minimum for the A->B->A dependency.
// ---------------------------------------------------------------------------
__global__ __launch_bounds__(128)
void ode_scan_kernel(const float* __restrict__ x,  const float* __restrict__ dt,
                     const float* __restrict__ W1, const float* __restrict__ b1,
                     const float* __restrict__ W2, const float* __restrict__ b2,
                     float* __restrict__ Z)
{
    __shared__ float sz[HID];
    __shared__ float sh[MID];

    const int  tid = threadIdx.x;
    const bool mid = (tid < MID);

    // Register-resident weight columns (one-time global loads).
    float w1r[HID + INP];
    #pragma unroll
    for (int k = 0; k < HID + INP; ++k) w1r[k] = mid ? W1[k * MID + tid] : 0.0f;
    const float b1r = mid ? b1[tid] : 0.0f;

    float w2r[MID];
    #pragma unroll
    for (int k = 0; k < MID; ++k) w2r[k] = W2[k * HID + tid];
    const float b2r = b2[tid];

    sz[tid] = 0.0f;
    __syncthreads();

    for (int t = 0; t < T_STEPS; ++t) {
        // Per-step inputs straight into registers (uniform addresses, cached);
        // reused across all 10 substeps -> no LDS broadcast, no extra barrier.
        float xr[INP];
        #pragma unroll
        for (int k = 0; k < INP; ++k) xr[k] = x[t * INP + k];
        const float hstep = (dt[2 * t + 1] - dt[2 * t]) * (1.0f / KSUB);

        for (int j = 0; j < KSUB; ++j) {
            // Stage A: h1 = tanh([z,x] @ W1 + b1); lanes 0..49 own one mid dim.
            if (mid) {
                float a0 = b1r, a1 = 0.f, a2 = 0.f, a3 = 0.f;
                #pragma unroll
                for (int k = 0; k < HID; k += 4) {
                    a0 = fmaf(sz[k + 0], w1r[k + 0], a0);
                    a1 = fmaf(sz[k + 1], w1r[k + 1], a1);
                    a2 = fmaf(sz[k + 2], w1r[k + 2], a2);
                    a3 = fmaf(sz[k + 3], w1r[k + 3], a3);
                }
                #pragma unroll
                for (int k = 0; k < INP; ++k)
                    a0 = fmaf(xr[k], w1r[HID + k], a0);
                sh[tid] = fast_tanh((a0 + a1) + (a2 + a3));
            }
            __syncthreads();
            // Stage B: z[tid] += hstep * tanh(h1 @ W2 + b2)[tid].
            {
                float a0 = b2r, a1 = 0.f;
                #pragma unroll
                for (int k = 0; k < MID; k += 2) {
                    a0 = fmaf(sh[k + 0], w2r[k + 0], a0);
                    a1 = fmaf(sh[k + 1], w2r[k + 1], a1);
                }
                sz[tid] = sz[tid] + hstep * fast_tanh(a0 + a1);
            }
            __syncthreads();
        }
        Z[t * HID + tid] = sz[tid];
    }
}

// ---------------------------------------------------------------------------
// Phase 2: batched heads over all 4096 timesteps, full-fp32 WMMA.
//   P0 = tanh(Z @ Wls1 + bls1); sigma = softplus(P0 . wls2 + bls2)
//   P1 = tanh(Z @ Wlm1 + blm1); mu    =          P1 . wls2 + bls2
// Block = 16 timesteps, 256 threads = 8 waves; wave w owns N-tile w (16 cols)
// for BOTH heads, sharing A-fragment loads. V_WMMA_F32_16X16X4_F32, K=128
// accumulated over 32 chunks of 4.
// ---------------------------------------------------------------------------
__global__ __launch_bounds__(256)
void heads_kernel(const float* __restrict__ Z,
                  const float* __restrict__ Wls1, const float* __restrict__ bls1,
                  const float* __restrict__ wls2, const float* __restrict__ bls2,
                  const float* __restrict__ Wlm1, const float* __restrict__ blm1,
                  float* __restrict__ out)
{
    __shared__ float sAcc[2][16];

    const int tid  = threadIdx.x;
    const int lane = tid & 31;
    const int wave = tid >> 5;        // 0..7 -> N tile
    const int half = lane >> 4;       // 0 or 1
    const int l16  = lane & 15;
    const int r0   = blockIdx.x * 16; // first timestep of this block

    if (tid < 32) ((float*)sAcc)[tid] = 0.0f;
    __syncthreads();

    const int    col  = wave * 16 + l16;        // output column 0..127
    const float* zrow = Z + (r0 + l16) * HID;   // A row for this lane

    v8f c0 = {};  // sigma head accumulator (Wls1)
    v8f c1 = {};  // mu head accumulator    (Wlm1)
    #pragma unroll 4
    for (int kk = 0; kk < 32; ++kk) {
        const int k = kk * 4 + half * 2;
        v2f a;  a.x  = zrow[k];              a.y  = zrow[k + 1];
        v2f bA; bA.x = Wls1[k * HID + col];  bA.y = Wls1[(k + 1) * HID + col];
        v2f bB; bB.x = Wlm1[k * HID + col];  bB.y = Wlm1[(k + 1) * HID + col];
        c0 = __builtin_amdgcn_wmma_f32_16x16x4_f32(false, a, false, bA,
                                                   (short)0, c0, false, false);
        c1 = __builtin_amdgcn_wmma_f32_16x16x4_f32(false, a, false, bB,
                                                   (short)0, c1, false, false);
    }

    const float bs = bls1[col];
    const float bm = blm1[col];
    const float w2 = wls2[col];

    #pragma unroll
    for (int i = 0; i < 8; ++i) {
        // C/D layout: VGPR i, lane L -> row (i + 8*(L>>4)), col (16*wave + L%16)
        float vs = fast_tanh(c0[i] + bs) * w2;
        float vm = fast_tanh(c1[i] + bm) * w2;
        // XOR-butterfly sum over the 16 lanes of this half (this N-tile's cols)
        vs += SWIZ_XOR(vs, 8);  vm += SWIZ_XOR(vm, 8);
        vs += SWIZ_XOR(vs, 4);  vm += SWIZ_XOR(vm, 4);
        vs += SWIZ_XOR(vs, 2);  vm += SWIZ_XOR(vm, 2);
        vs += SWIZ_XOR(vs, 1);  vm += SWIZ_XOR(vm, 1);
        if (l16 == 0) {
            const int row = i + half * 8;
            atomicAdd(&sAcc[0][row], vs);
            atomicAdd(&sAcc[1][row], vm);
        }
    }
    __syncthreads();

    if (tid < 16) {
        const float c  = bls2[0];
        const float mu = sAcc[1][tid] + c;
        const float sp = sAcc[0][tid] + c;
        const float sig = (sp > 0.0f) ? (sp + log1pf(expf(-sp)))
                                      : log1pf(expf(sp));
        out[r0 + tid]           = mu;   // mu block:    [0, T)
        out[T_STEPS + r0 + tid] = sig;  // sigma block: [T, 2T)
    }
}

// ---------------------------------------------------------------------------
extern "C" void kernel_launch(void* const* d_in, const int* in_sizes, int n_in,
                              void* d_out, int out_size, void* d_ws, size_t ws_size,
                              hipStream_t stream)
{
    (void)in_sizes; (void)n_in; (void)out_size; (void)ws_size;
    const float* x    = (const float*)d_in[0];
    const float* dt   = (const float*)d_in[1];
    const float* W1   = (const float*)d_in[2];
    const float* b1   = (const float*)d_in[3];
    const float* W2   = (const float*)d_in[4];
    const float* b2   = (const float*)d_in[5];
    const float* Wls1 = (const float*)d_in[6];
    const float* bls1 = (const float*)d_in[7];
    const float* wls2 = (const float*)d_in[8];
    const float* bls2 = (const float*)d_in[9];
    const float* Wlm1 = (const float*)d_in[10];
    const float* blm1 = (const float*)d_in[11];

    float* Z = (float*)d_ws;  // T*HID floats = 2 MB of z_t states

    ode_scan_kernel<<<1, 128, 0, stream>>>(x, dt, W1, b1, W2, b2, Z);
    heads_kernel<<<T_STEPS / 16, 256, 0, stream>>>(Z, Wls1, bls1, wls2, bls2,
                                                   Wlm1, blm1, (float*)d_out);
}